// LocalConnectivity_23313082482755
// MI455X (gfx1250) — compile-verified
//
#include <hip/hip_runtime.h>
#include <stdint.h>

// Problem constants (match reference: B=64, H=512, W=512, MAX_DISTANCE=5)
#define BB   64
#define HH   512
#define WW   512
#define HALO 5

// Tile config: 128 threads (4 wave32), each thread owns one column and
// streams TY output rows. Block computes TX x TY outputs of one image.
#define TX 128
#define TY 32
#define LROWS   (TY + 2 * HALO)   // 42 staged rows
// Staged columns: re-based to a 16B-aligned origin (global col c0-8) so the
// fill can use b128 async loads. 144 cols cover c0-8 .. c0+135; lane c's taps
// are LDS cols c+3 .. c+13 (max 140), all in range.
#define LCOLS   144
#define LSTRIDE 144               // floats; row base 16B-aligned
#define COFF    8                 // LDS col 0 == global col (c0 - COFF)
#define NV4     (LCOLS / 4)       // 36 float4 groups per staged row

// CDNA5 async global->LDS path (ASYNCcnt-tracked), tiered guards so the file
// always compiles.
#if defined(__HIP_DEVICE_COMPILE__) && __has_builtin(__builtin_amdgcn_s_wait_asynccnt)
  #if __has_builtin(__builtin_amdgcn_global_load_async_to_lds_b128)
    #define ASYNC_TIER 2
  #elif __has_builtin(__builtin_amdgcn_global_load_async_to_lds_b32)
    #define ASYNC_TIER 1
  #else
    #define ASYNC_TIER 0
  #endif
#else
  #define ASYNC_TIER 0
#endif

typedef int v4i __attribute__((vector_size(16)));
typedef __attribute__((address_space(1))) int g_int;
typedef __attribute__((address_space(3))) int l_int;
typedef __attribute__((address_space(1))) v4i g_v4i;
typedef __attribute__((address_space(3))) v4i l_v4i;

__global__ __launch_bounds__(TX) void
LocalConnectivity_23313082482755_kernel(const float* __restrict__ in,
                                        const float* __restrict__ wts,
                                        float* __restrict__ out) {
    __shared__ float smem[LROWS * LSTRIDE];

    const int c  = threadIdx.x;        // 0..127, one column per thread
    const int c0 = blockIdx.x * TX;    // tile origin (cols)
    const int r0 = blockIdx.y * TY;    // tile origin (rows)
    const int b  = blockIdx.z;         // batch image

    const float* __restrict__ img = in + (size_t)b * (HH * WW);

    // ---- Stage halo tile (42 x 144 floats) into LDS, torus-wrapped ----
#if ASYNC_TIER == 2
    // float4-granular staging: wrap at group granularity is exact (512 % 4 == 0).
    const int g4base = (c0 >> 2) - (COFF >> 2);          // first float4 group (pre-wrap)
    for (int idx = c; idx < LROWS * NV4; idx += TX) {
        const int gy   = idx / NV4;
        const int i4   = idx - gy * NV4;
        const int grow = (r0 + gy - HALO) & (HH - 1);
        const int g4   = (g4base + i4) & ((WW >> 2) - 1);
        __builtin_amdgcn_global_load_async_to_lds_b128(
            (g_v4i*)(img + (size_t)grow * WW + (g4 << 2)),
            (l_v4i*)(&smem[gy * LSTRIDE + (i4 << 2)]),
            /*offset=*/0, /*cpol=*/0);
    }
    __builtin_amdgcn_s_wait_asynccnt(0);
#else
    for (int idx = c; idx < LROWS * LCOLS; idx += TX) {
        const int gy   = idx / LCOLS;
        const int gx   = idx - gy * LCOLS;
        const int grow = (r0 + gy - HALO) & (HH - 1);
        const int gcol = (c0 + gx - COFF) & (WW - 1);
#if ASYNC_TIER == 1
        __builtin_amdgcn_global_load_async_to_lds_b32(
            (g_int*)(img + (size_t)grow * WW + gcol),
            (l_int*)(&smem[gy * LSTRIDE + gx]),
            /*offset=*/0, /*cpol=*/0);
#else
        smem[gy * LSTRIDE + gx] = img[(size_t)grow * WW + gcol];
#endif
    }
#if ASYNC_TIER == 1
    __builtin_amdgcn_s_wait_asynccnt(0);
#endif
#endif
    __syncthreads();

    const float w1 = wts[0], w2 = wts[1], w3 = wts[2], w4 = wts[3], w5 = wts[4];

    float acc[TY];
#pragma unroll
    for (int o = 0; o < TY; ++o) acc[o] = 0.0f;

    // ---- Factored diamond stencil ----
    // Per source row: a_k = p[-k] + p[+k]  (horizontal symmetric sums)
    //                 P_m = sum_k w_{m+k} * a_k  (P0 excludes the center tap)
    // Row at vertical offset m from an output contributes P_{|m|} to it.
#pragma unroll
    for (int rr = 0; rr < LROWS; ++rr) {
        const float* p = &smem[rr * LSTRIDE + (c + COFF)];
        const float a0 = p[0];
        const float a1 = p[-1] + p[1];
        const float a2 = p[-2] + p[2];
        const float a3 = p[-3] + p[3];
        const float a4 = p[-4] + p[4];
        const float a5 = p[-5] + p[5];

        float Pm[6];
        Pm[0] = w1 * a1 + w2 * a2 + w3 * a3 + w4 * a4 + w5 * a5;  // center row, no (0,0)
        Pm[1] = w1 * a0 + w2 * a1 + w3 * a2 + w4 * a3 + w5 * a4;
        Pm[2] = w2 * a0 + w3 * a1 + w4 * a2 + w5 * a3;
        Pm[3] = w3 * a0 + w4 * a1 + w5 * a2;
        Pm[4] = w4 * a0 + w5 * a1;
        Pm[5] = w5 * a0;

#pragma unroll
        for (int o = 0; o < TY; ++o) {
            const int m  = rr - HALO - o;          // source-row offset vs output o
            const int am = (m < 0) ? -m : m;
            if (am <= HALO) acc[o] += Pm[am];      // folds to constant-index adds
        }
    }

    // ---- Coalesced stores: lane c -> column c0+c, rows r0..r0+TY-1 ----
    float* __restrict__ orow =
        out + (size_t)b * (HH * WW) + (size_t)r0 * WW + (c0 + c);
#pragma unroll
    for (int o = 0; o < TY; ++o) orow[(size_t)o * WW] = acc[o];
}

extern "C" void kernel_launch(void* const* d_in, const int* in_sizes, int n_in,
                              void* d_out, int out_size, void* d_ws, size_t ws_size,
                              hipStream_t stream) {
    (void)in_sizes; (void)n_in; (void)d_ws; (void)ws_size; (void)out_size;
    const float* grid_spikes      = (const float*)d_in[0];  // [64,512,512] f32
    const float* distance_weights = (const float*)d_in[1];  // [5] f32
    float* out = (float*)d_out;                             // [64,512,512] f32

    dim3 grid(WW / TX, HH / TY, BB);   // (4, 16, 64) = 4096 blocks
    dim3 block(TX);                    // 128 threads = 4 wave32
    LocalConnectivity_23313082482755_kernel<<<grid, block, 0, stream>>>(
        grid_spikes, distance_weights, out);
}